// CombinedKSpaceRowwiseMSELoss_35373350650399
// MI455X (gfx1250) — compile-verified
//
#include <hip/hip_runtime.h>

// CombinedKSpaceRowwiseMSELoss for (32, 2, 640, 640) f32 pred/target.
// out = sum_sq_central / N_c + sum_abs_periphery / N_p  (both channels pooled;
// all weights == 1 and both channels share normalizers).
//
// Phase 1: bandwidth-bound streaming partial reduction (non-temporal b128
//          loads; ~210 MB single pass -> ~9 us at 23.3 TB/s).
// Phase 2: single-wave finalize using V_WMMA_F32_16X16X4_F32 with B=ones as an
//          exact-fp32 64-way summation tree (D = A x 1 + C accumulates row
//          sums). nblocks is host-guaranteed to be a multiple of 64 so the
//          loop needs no predication: EXEC stays all-1s (WMMA requirement)
//          and each lane fetches its two (sq, ab) pairs with one b128 load.

typedef __attribute__((ext_vector_type(2))) float v2f;
typedef __attribute__((ext_vector_type(4))) float v4f;
typedef __attribute__((ext_vector_type(8))) float v8f;

#define CHUNKS_PER_ROW 160   // 640 / 4
#define CH_START        60   // 240 / 4
#define CH_END         100   // 400 / 4
#define INV_NC (1.0f / 3276800.0f)   // 32*640*160
#define INV_NP (1.0f / 9830400.0f)   // 32*640*480

__global__ __launch_bounds__(256)
void kspace_partial_kernel(const float* __restrict__ pred,
                           const float* __restrict__ targ,
                           float* __restrict__ partials,
                           unsigned n4) {
    const unsigned tid    = threadIdx.x;
    const unsigned stride = gridDim.x * blockDim.x;
    const v4f* __restrict__ p4 = (const v4f*)pred;
    const v4f* __restrict__ t4 = (const v4f*)targ;

    float acc_sq = 0.0f;   // central: sum of diff^2
    float acc_ab = 0.0f;   // periphery: sum of |diff|

#pragma unroll 4
    for (unsigned i = blockIdx.x * blockDim.x + tid; i < n4; i += stride) {
        v4f p = __builtin_nontemporal_load(p4 + i);   // global_load_b128, TH=NT
        v4f t = __builtin_nontemporal_load(t4 + i);
        v4f d = p - t;
        unsigned cw = i % CHUNKS_PER_ROW;             // 4-chunk column within a row
        bool central = (cw >= CH_START) & (cw < CH_END);
        float sq = d.x * d.x + d.y * d.y + d.z * d.z + d.w * d.w;
        float ab = __builtin_fabsf(d.x) + __builtin_fabsf(d.y) +
                   __builtin_fabsf(d.z) + __builtin_fabsf(d.w);
        acc_sq += central ? sq : 0.0f;
        acc_ab += central ? 0.0f : ab;
    }

    __shared__ float s_sq[256];
    __shared__ float s_ab[256];
    s_sq[tid] = acc_sq;
    s_ab[tid] = acc_ab;
    __syncthreads();
    for (unsigned s = 128; s > 0; s >>= 1) {
        if (tid < s) {
            s_sq[tid] += s_sq[tid + s];
            s_ab[tid] += s_ab[tid + s];
        }
        __syncthreads();
    }
    if (tid == 0) {
        partials[2u * blockIdx.x]     = s_sq[0];
        partials[2u * blockIdx.x + 1] = s_ab[0];
    }
}

// One wave32. nblocks is a multiple of 64. Each iteration: lane L loads the
// v4f holding pairs (it*64 + 2L) and (it*64 + 2L + 1) = {sq0, ab0, sq1, ab1},
// feeding two v_wmma_f32_16x16x4_f32 accumulations (sq stream, ab stream).
// C/D layout: VGPR r holds D[r][lane] (lanes 0-15) and D[8+r][lane-16]
// (lanes 16-31); in-lane sum of the 8 C VGPRs + one xor-16 shuffle yields the
// grand total.
__global__ __launch_bounds__(32)
void kspace_final_kernel(const float* __restrict__ partials,
                         int nblocks,
                         float* __restrict__ out) {
    const int lane = threadIdx.x;

    v8f c_sq = {};
    v8f c_ab = {};
    v2f ones;
    ones.x = 1.0f;
    ones.y = 1.0f;

    const v4f* __restrict__ pp = (const v4f*)partials;  // one v4f = 2 (sq,ab) pairs
    const int iters = nblocks >> 6;                     // 64 pairs per iteration
    for (int it = 0; it < iters; ++it) {
        v4f f = pp[it * 32 + lane];
        v2f a_sq, a_ab;
        a_sq.x = f.x;  a_sq.y = f.z;
        a_ab.x = f.y;  a_ab.y = f.w;
        // v_wmma_f32_16x16x4_f32: D = A x B + C (exact fp32 MAC path)
        c_sq = __builtin_amdgcn_wmma_f32_16x16x4_f32(
            false, a_sq, false, ones, (short)0, c_sq, false, false);
        c_ab = __builtin_amdgcn_wmma_f32_16x16x4_f32(
            false, a_ab, false, ones, (short)0, c_ab, false, false);
    }

    float s_sq = ((c_sq[0] + c_sq[1]) + (c_sq[2] + c_sq[3])) +
                 ((c_sq[4] + c_sq[5]) + (c_sq[6] + c_sq[7]));
    float s_ab = ((c_ab[0] + c_ab[1]) + (c_ab[2] + c_ab[3])) +
                 ((c_ab[4] + c_ab[5]) + (c_ab[6] + c_ab[7]));
    // combine row groups 0-7 (lanes 0-15) with 8-15 (lanes 16-31)
    s_sq += __shfl_xor(s_sq, 16, 32);
    s_ab += __shfl_xor(s_ab, 16, 32);

    if (lane == 0) {
        out[0] = s_sq * INV_NC + s_ab * INV_NP;
    }
}

extern "C" void kernel_launch(void* const* d_in, const int* in_sizes, int n_in,
                              void* d_out, int out_size, void* d_ws, size_t ws_size,
                              hipStream_t stream) {
    const float* pred = (const float*)d_in[0];
    const float* targ = (const float*)d_in[1];
    float* out      = (float*)d_out;
    float* partials = (float*)d_ws;

    const unsigned n  = (unsigned)in_sizes[0];   // 26,214,400
    const unsigned n4 = n / 4u;                  // 6,553,600 float4 chunks

    // nblocks must be a multiple of 64 (finalize kernel invariant).
    int nblocks = 1024;                          // 8 KB of ws for partial pairs
    if (ws_size < (size_t)nblocks * 2u * sizeof(float)) {
        nblocks = (int)(ws_size / (2u * sizeof(float))) & ~63;
        if (nblocks < 64) nblocks = 64;          // ws is harness-preallocated; >=512B assumed
    }

    kspace_partial_kernel<<<nblocks, 256, 0, stream>>>(pred, targ, partials, n4);
    kspace_final_kernel<<<1, 32, 0, stream>>>(partials, nblocks, out);
}